// PerspectiveDilatedConv2dLayer_68994354642974
// MI455X (gfx1250) — compile-verified
//
#include <hip/hip_runtime.h>
#include <hip/hip_bf16.h>

// Perspective dilated conv2d (B=4, Cin=64, Cout=64, K=3, H=W=160, stride=1)
// as an im2col GEMM on the CDNA5 fp32 WMMA path (v_wmma_f32_16x16x4_f32),
// with the weight panel staged into LDS by the Tensor Data Mover
// (tensor_load_to_lds, double-buffered, TENSORcnt-synchronized).

typedef float v2f  __attribute__((ext_vector_type(2)));
typedef float v8f  __attribute__((ext_vector_type(8)));
typedef unsigned int u32x4 __attribute__((ext_vector_type(4)));
typedef int   i32x4 __attribute__((ext_vector_type(4)));
typedef int   i32x8 __attribute__((ext_vector_type(8)));

#define IMG_W   160
#define IMG_HW  25600          // 160*160
#define CIN     64
#define COUT    64
#define KDIM    576            // Cin*3*3
#define PIX_PER_WG 32
#define NSAMP   288            // 32 pixels * 9 taps
#define KC      36             // K-chunk: 4 channels * 9 taps
#define NCHUNK  16             // 576 / 36
#define APITCH  42             // sA row pitch (floats): even + conflict-free
#define BPITCH  36             // sB row pitch: TDM writes tile rows contiguously

#if defined(__has_builtin)
#if __has_builtin(__builtin_amdgcn_tensor_load_to_lds) && \
    __has_builtin(__builtin_amdgcn_s_wait_tensorcnt)
#define HAVE_TDM 1
#endif
#endif

#if defined(HAVE_TDM)
#if __has_include(<hip/amd_detail/amd_gfx1250_TDM.h>)
// amdgpu-toolchain (clang-23 / therock-10.0 headers): 6-arg builtin
#define TDM_LOAD(g0, g1, g2, g3, z8) \
    __builtin_amdgcn_tensor_load_to_lds((g0), (g1), (g2), (g3), (z8), 0)
#else
// ROCm 7.2 (clang-22): 5-arg builtin
#define TDM_LOAD(g0, g1, g2, g3, z8) \
    __builtin_amdgcn_tensor_load_to_lds((g0), (g1), (g2), (g3), 0)
#endif
#endif

__global__ __launch_bounds__(256)
void pdconv2d_wmma_f32_kernel(const float* __restrict__ in,
                              const float* __restrict__ wgt,
                              const float* __restrict__ rate,
                              const float* __restrict__ bias,
                              float* __restrict__ out)
{
    __shared__ int   sIdx[NSAMP][4];          // 4 corner linear indices per (pixel,tap)
    __shared__ float sCW [NSAMP][4];          // 4 bilinear corner weights (0 if OOB)
    __shared__ float sA[PIX_PER_WG][APITCH];  // im2col panel: [pixel][k-in-chunk]
    __shared__ float sB[2][COUT][BPITCH];     // weight panel (double-buffered): [n][k]

    const int tid = threadIdx.x;
    const int wg  = blockIdx.x;

    // 32 consecutive pixels of one row of one image (160 % 32 == 0)
    const int pixel_base = wg * PIX_PER_WG;
    const int b   = pixel_base / IMG_HW;
    const int rem = pixel_base - b * IMG_HW;
    const int h   = rem / IMG_W;
    const int w0  = rem - h * IMG_W;

    const int wave = tid >> 5;
    const int lane = tid & 31;
    const int mt   = wave & 1;      // which 16-pixel half
    const int nt   = wave >> 1;     // which 16-channel quarter
    const int lo   = lane & 15;
    const int hi   = lane >> 4;

#if defined(HAVE_TDM)
    // ---- TDM descriptor (ISA 8.3/8.4): 2D tile 36x64 f32 out of the 64x576
    // weight matrix, row stride 576 dwords, written contiguously into sB[buf].
    u32x4 g0;
    i32x8 g1;
    i32x4 gz = {0, 0, 0, 0};
    i32x8 z8 = {0, 0, 0, 0, 0, 0, 0, 0};
    const unsigned ldsB0 = (unsigned)(size_t)&sB[0][0][0];   // LDS byte offset
    const unsigned ldsB1 = (unsigned)(size_t)&sB[1][0][0];
    const unsigned long long wbase = (unsigned long long)(size_t)wgt;

    g1[0] = 0x00020000;                                  // data_size = 4B
    g1[1] = (int)((unsigned)KDIM << 16);                 // tensor_dim0[15:0]
    g1[2] = (int)(((unsigned)KDIM >> 16) | ((unsigned)COUT << 16)); // dim0 hi | dim1 lo
    g1[3] = (int)((unsigned)KC << 16);                   // dim1 hi | tile_dim0
    g1[4] = COUT;                                        // tile_dim1 (tile_dim2 = 0)
    g1[5] = KDIM;                                        // tensor_dim0_stride[31:0]
    g1[6] = 0;                                           // stride hi | dim1_stride lo
    g1[7] = 0;

    g0.x = 1u;                                           // count = 1 (valid D#)
    if (wave == 0) {                                     // prefetch chunk 0 -> sB[0]
        g0.y = ldsB0;
        g0.z = (unsigned)wbase;
        g0.w = ((unsigned)(wbase >> 32) & 0x01FFFFFFu) | 0x80000000u; // addr hi | type=2
        TDM_LOAD(g0, g1, gz, gz, z8);
    }
#endif

    // ---- Stage 0: per-(pixel,tap) bilinear corner precompute (shared by all channels)
    for (int i = tid; i < NSAMP; i += 256) {
        const int p  = i / 9;
        const int t  = i - p * 9;
        const int ky = t / 3;
        const int kx = t - ky * 3;
        const float r  = rate[b * IMG_HW + h * IMG_W + w0 + p];
        const float sy = (float)h        + (float)(ky - 1) * r;
        const float sx = (float)(w0 + p) + (float)(kx - 1) * r;
        const float y0f = floorf(sy), x0f = floorf(sx);
        const float wy = sy - y0f, wx = sx - x0f;
        const int y0 = (int)y0f, x0 = (int)x0f;
#pragma unroll
        for (int c4 = 0; c4 < 4; ++c4) {
            const int dy = c4 >> 1, dx = c4 & 1;
            const int yy = y0 + dy, xx = x0 + dx;
            const bool valid = (yy >= 0) && (yy < IMG_W) && (xx >= 0) && (xx < IMG_W);
            const int yc = min(max(yy, 0), IMG_W - 1);
            const int xc = min(max(xx, 0), IMG_W - 1);
            const float cw = (dy ? wy : 1.0f - wy) * (dx ? wx : 1.0f - wx);
            sIdx[i][c4] = yc * IMG_W + xc;
            sCW [i][c4] = valid ? cw : 0.0f;
        }
    }

    // C/D layout (ISA 7.12.2): VGPR r <-> row (r + 8*hi), col lo. Init from bias.
    v8f acc;
    {
        const float bv = bias[nt * 16 + lo];
#pragma unroll
        for (int r = 0; r < 8; ++r) acc[r] = bv;
    }

    const float* inB = in + (size_t)b * (CIN * IMG_HW);

    // ---- Main loop: 16 channel-groups of 4 channels (K chunk = 36)
    for (int cg = 0; cg < NCHUNK; ++cg) {
        __syncthreads();  // corners ready (iter 0) / previous chunk fully consumed

        // Build A panel: sA[p][cc*9+t] = bilinear sample of channel (cg*4+cc), tap t
        for (int i = tid; i < NSAMP; i += 256) {
            const int p = i / 9;
            const int t = i - p * 9;
            const int i0 = sIdx[i][0], i1 = sIdx[i][1], i2 = sIdx[i][2], i3 = sIdx[i][3];
            const float c0 = sCW[i][0], c1 = sCW[i][1], c2 = sCW[i][2], c3 = sCW[i][3];
#pragma unroll
            for (int cc = 0; cc < 4; ++cc) {
                const float* base = inB + (size_t)(cg * 4 + cc) * IMG_HW;
                const float v = c0 * base[i0] + c1 * base[i1]
                              + c2 * base[i2] + c3 * base[i3];
                sA[p][cc * 9 + t] = v;
            }
        }

#if defined(HAVE_TDM)
        // Wave 0: kick the DMA for the *next* chunk, then guarantee the current
        // chunk's (older, in-order) DMA has landed before signaling the barrier.
        if (wave == 0) {
            if (cg + 1 < NCHUNK) {
                const unsigned long long ga =
                    wbase + (unsigned long long)(cg + 1) * (KC * 4);
                g0.y = ((cg + 1) & 1) ? ldsB1 : ldsB0;
                g0.z = (unsigned)ga;
                g0.w = ((unsigned)(ga >> 32) & 0x01FFFFFFu) | 0x80000000u;
                TDM_LOAD(g0, g1, gz, gz, z8);
                __builtin_amdgcn_s_wait_tensorcnt(1);
            } else {
                __builtin_amdgcn_s_wait_tensorcnt(0);
            }
        }
#else
        // Fallback: manual weight-panel copy (chunk k-slice is contiguous per row)
        for (int i = tid; i < COUT * KC; i += 256) {
            const int n = i / KC;
            const int k = i - n * KC;
            sB[cg & 1][n][k] = wgt[n * KDIM + cg * KC + k];
        }
#endif
        __syncthreads();

        // 9 fp32 WMMAs per wave per chunk (K=36 / 4).
        // A lanes 0-15 hold K={kk,kk+1}, lanes 16-31 hold K={kk+2,kk+3};
        // B mirrors it with column = lo. Both are one aligned v2f LDS read.
        const float (*sBc)[BPITCH] = sB[cg & 1];
#pragma unroll
        for (int kk = 0; kk < KC; kk += 4) {
            const int kl = kk + 2 * hi;
            const v2f a  = *(const v2f*)&sA[mt * 16 + lo][kl];
            const v2f bb = *(const v2f*)&sBc[nt * 16 + lo][kl];
            acc = __builtin_amdgcn_wmma_f32_16x16x4_f32(
                      /*neg_a=*/false, a, /*neg_b=*/false, bb,
                      /*c_mod=*/(short)0, acc,
                      /*reuse_a=*/false, /*reuse_b=*/false);
        }
    }

    // ---- Store: out[b][n][h][w], lane's column n = nt*16+lo, rows -> consecutive w
    float* outp = out + ((size_t)b * COUT + nt * 16 + lo) * IMG_HW
                      + h * IMG_W + w0 + mt * 16;
#pragma unroll
    for (int r = 0; r < 8; ++r) {
        outp[r + 8 * hi] = acc[r];
    }
}

extern "C" void kernel_launch(void* const* d_in, const int* in_sizes, int n_in,
                              void* d_out, int out_size, void* d_ws, size_t ws_size,
                              hipStream_t stream) {
    const float* in   = (const float*)d_in[0];   // (4,64,160,160) f32
    const float* wgt  = (const float*)d_in[1];   // (64,64,3,3)   f32
    const float* rate = (const float*)d_in[2];   // (4,1,160,160) f32
    const float* bias = (const float*)d_in[3];   // (64,)         f32
    // d_in[4], d_in[5]: stride_h / stride_w == 1 (hardcoded)

    float* out = (float*)d_out;                  // (4,64,160,160) f32

    const int total_pixels = 4 * IMG_HW;         // 102400
    const int blocks = total_pixels / PIX_PER_WG;// 3200
    pdconv2d_wmma_f32_kernel<<<blocks, 256, 0, stream>>>(in, wgt, rate, bias, out);
}